// GNN_new_37048387895328
// MI455X (gfx1250) — compile-verified
//
#include <hip/hip_runtime.h>

// ---------------- problem constants ----------------
#define B_SZ    8
#define N_NODE  128
#define D_DIM   512
#define K_REG   36
#define N_TOT   (B_SZ * N_NODE)      // 1024
#define ROWS_GS (N_TOT * K_REG)      // 36864

typedef float v2f __attribute__((ext_vector_type(2)));
typedef float v8f __attribute__((ext_vector_type(8)));

enum { EPI_NONE = 0, EPI_BIAS = 1, EPI_BIAS_RELU = 2, EPI_BIAS_RELU_MUL = 3, EPI_LOGITS = 4 };

// Generic fp32 WMMA GEMM: C[M,N] = A[M,Kd] @ B[Kd,N] (+ epilogue).
// Block = 128 threads (4 waves); block tile = 128 rows x 64 cols.
// Wave tile = 32 rows x 64 cols: 2 A-fragments x 4 B-fragments -> 8 accumulators,
// so each B fragment is reused twice (halves B read traffic vs 16-row tiles).
// All flat indices fit in 32 bits -> uniform scalar base + 32-bit voffsets (saddr form).
template <int EPI>
__global__ __launch_bounds__(128) void gemm_wmma_f32(
    const float* __restrict__ A, const float* __restrict__ Bm, float* __restrict__ C,
    int M, int N, int Kd,
    long strideA, long strideB, long strideC,          // batch strides (grid.z)
    const float* __restrict__ bias, float bias_scale,  // EPI_BIAS*: per-col bias
    const float* __restrict__ ep0,                     // LOGITS: qhi[N_TOT,D] ; MUL: q_repr
    const float* __restrict__ ep1,                     // LOGITS: Wl[D]
    float* __restrict__ logits)                        // LOGITS: [ROWS_GS] (pre-init to bl)
{
  const int b = blockIdx.z;
  A  += (long)b * strideA;   // uniform (SALU) batch adjust; bases stay scalar
  Bm += (long)b * strideB;
  const int wave = threadIdx.x >> 5;
  const int lane = threadIdx.x & 31;
  const int half = lane >> 4;   // 0: lanes 0-15, 1: lanes 16-31
  const int l16  = lane & 15;
  const int mbase = blockIdx.y * 128 + wave * 32;
  const int nbase = blockIdx.x * 64;

  // 32-bit per-lane offsets (elements). A VGPR0 holds K=k0 (lo lanes) / k0+2 (hi lanes).
  int aoff0 = (mbase + l16) * Kd + 2 * half;           // rows mbase..mbase+15
  int aoff1 = aoff0 + 16 * Kd;                         // rows mbase+16..mbase+31
  int boff  = (2 * half) * N + nbase + l16;            // B row kk = k0 + 2*half

  v8f acc[2][4] = {};

  for (int k0 = 0; k0 < Kd; k0 += 4) {
    v2f a0 = *(const v2f*)(A + aoff0 + k0);            // one b64: A[m][kk..kk+1]
    v2f a1 = *(const v2f*)(A + aoff1 + k0);
    const int bk = boff + k0 * N;
    v2f bf[4];
#pragma unroll
    for (int t = 0; t < 4; t++) {
      bf[t].x = Bm[bk + 16 * t];                       // B[kk][col]
      bf[t].y = Bm[bk + N + 16 * t];                   // B[kk+1][col]
    }
#pragma unroll
    for (int t = 0; t < 4; t++) {
      acc[0][t] = __builtin_amdgcn_wmma_f32_16x16x4_f32(false, a0, false, bf[t], (short)0, acc[0][t], false, false);
      acc[1][t] = __builtin_amdgcn_wmma_f32_16x16x4_f32(false, a1, false, bf[t], (short)0, acc[1][t], false, false);
    }
  }

  // D layout: acc[s][t][r] = C[mbase + 16*s + r + 8*half][nbase + 16*t + l16]
  if (EPI == EPI_LOGITS) {
    // partial logit per row: sum_col relu(acc + qhi[n,col]) * Wl[col], n = row/K_REG
#pragma unroll
    for (int s = 0; s < 2; s++) {
#pragma unroll
      for (int r = 0; r < 8; r++) {
        const int row = mbase + 16 * s + r + 8 * half;
        const int n = row / K_REG;
        float partial = 0.f;
#pragma unroll
        for (int t = 0; t < 4; t++) {
          const int col = nbase + 16 * t + l16;
          float v = acc[s][t][r] + ep0[n * D_DIM + col];
          v = fmaxf(v, 0.f);
          partial += v * ep1[col];
        }
        // reduce across the 16-lane half (xor masks < 16 never cross halves)
        for (int off = 8; off >= 1; off >>= 1) partial += __shfl_xor(partial, off, 32);
        if (l16 == 0) atomicAdd(&logits[row], partial);
      }
    }
  } else {
    C += (long)b * strideC;
#pragma unroll
    for (int s = 0; s < 2; s++) {
#pragma unroll
      for (int t = 0; t < 4; t++) {
#pragma unroll
        for (int r = 0; r < 8; r++) {
          const int row = mbase + 16 * s + r + 8 * half;
          const int col = nbase + 16 * t + l16;
          float v = acc[s][t][r];
          if (EPI == EPI_BIAS) v += bias_scale * bias[col];
          if (EPI == EPI_BIAS_RELU || EPI == EPI_BIAS_RELU_MUL) v = fmaxf(v + bias[col], 0.f);
          if (EPI == EPI_BIAS_RELU_MUL) v *= ep0[row * N + col];
          C[row * N + col] = v;
        }
      }
    }
  }
}

__global__ __launch_bounds__(256) void init_logits(float* __restrict__ logits,
                                                   const float* __restrict__ bl) {
  int i = blockIdx.x * 256 + threadIdx.x;
  if (i < ROWS_GS) logits[i] = bl[0];
}

// One block per node: softmax over K_REG logits, then v_emb = sum_k att_k * gs[n,k,:]
__global__ __launch_bounds__(256) void softmax_vemb(const float* __restrict__ logits,
                                                    const float* __restrict__ gs,
                                                    float* __restrict__ v_emb) {
  __shared__ float att[K_REG];
  const int n = blockIdx.x;
  const int tid = threadIdx.x;
  if (tid < K_REG) att[tid] = logits[n * K_REG + tid];
  __syncthreads();
  if (tid == 0) {  // K_REG=36: serial softmax is negligible
    float m = -1e30f;
    for (int k = 0; k < K_REG; k++) m = fmaxf(m, att[k]);
    float s = 0.f;
    for (int k = 0; k < K_REG; k++) { float e = expf(att[k] - m); att[k] = e; s += e; }
    float inv = 1.f / s;
    for (int k = 0; k < K_REG; k++) att[k] *= inv;
  }
  __syncthreads();
  const float* g = gs + (long)n * K_REG * D_DIM;
  for (int d = tid; d < D_DIM; d += 256) {
    float s = 0.f;
#pragma unroll 4
    for (int k = 0; k < K_REG; k++) s += att[k] * g[k * D_DIM + d];
    v_emb[n * D_DIM + d] = s;
  }
}

extern "C" void kernel_launch(void* const* d_in, const int* in_sizes, int n_in,
                              void* d_out, int out_size, void* d_ws, size_t ws_size,
                              hipStream_t stream) {
  const float* cur  = (const float*)d_in[0];   // [N_TOT, D]
  const float* mask = (const float*)d_in[1];   // [B, 128, 128]
  const float* gs   = (const float*)d_in[2];   // [ROWS_GS, D]
  const float* Wp   = (const float*)d_in[3];   // [D, D]
  const float* bp   = (const float*)d_in[4];
  const float* Wn   = (const float*)d_in[5];   // [2D, D]  rows 0..511 = Wn_lo, 512..1023 = Wn_hi
  const float* bn   = (const float*)d_in[6];
  const float* Wl   = (const float*)d_in[7];   // [D, 1]
  const float* bl   = (const float*)d_in[8];
  const float* Wv   = (const float*)d_in[9];
  const float* bv   = (const float*)d_in[10];
  const float* Wq   = (const float*)d_in[11];
  const float* bq   = (const float*)d_in[12];

  float* ws = (float*)d_ws;
  float* agg    = ws;                         // 1024*512
  float* nbrs   = agg    + (long)N_TOT * D_DIM;
  float* qhi    = nbrs   + (long)N_TOT * D_DIM;
  float* qrep   = qhi    + (long)N_TOT * D_DIM;
  float* logits = qrep   + (long)N_TOT * D_DIM;   // 36864
  float* vemb   = logits + ROWS_GS;

  const dim3 blk(128);

  // 1) agg[b] = mask[b] (128x128) @ h[b] (128x512)
  gemm_wmma_f32<EPI_NONE><<<dim3(D_DIM / 64, N_NODE / 128, B_SZ), blk, 0, stream>>>(
      mask, cur, agg, N_NODE, D_DIM, N_NODE,
      (long)N_NODE * N_NODE, (long)N_NODE * D_DIM, (long)N_NODE * D_DIM,
      nullptr, 0.f, nullptr, nullptr, nullptr);

  // 2) neighbours = agg @ Wp + N_NODE * bp
  gemm_wmma_f32<EPI_BIAS><<<dim3(D_DIM / 64, N_TOT / 128, 1), blk, 0, stream>>>(
      agg, Wp, nbrs, N_TOT, D_DIM, D_DIM, 0, 0, 0,
      bp, (float)N_NODE, nullptr, nullptr, nullptr);

  // 3a) qhi = neighbours @ Wn_hi + bn   (shared across all K regions per node)
  gemm_wmma_f32<EPI_BIAS><<<dim3(D_DIM / 64, N_TOT / 128, 1), blk, 0, stream>>>(
      nbrs, Wn + (long)D_DIM * D_DIM, qhi, N_TOT, D_DIM, D_DIM, 0, 0, 0,
      bn, 1.f, nullptr, nullptr, nullptr);

  // 3b) q_repr = relu(neighbours @ Wq + bq)
  gemm_wmma_f32<EPI_BIAS_RELU><<<dim3(D_DIM / 64, N_TOT / 128, 1), blk, 0, stream>>>(
      nbrs, Wq, qrep, N_TOT, D_DIM, D_DIM, 0, 0, 0,
      bq, 1.f, nullptr, nullptr, nullptr);

  // 4) logits = Wl . relu(gs @ Wn_lo + qhi) + bl  (joint never materialized)
  init_logits<<<(ROWS_GS + 255) / 256, 256, 0, stream>>>(logits, bl);
  gemm_wmma_f32<EPI_LOGITS><<<dim3(D_DIM / 64, ROWS_GS / 128, 1), blk, 0, stream>>>(
      gs, Wn, nullptr, ROWS_GS, D_DIM, D_DIM, 0, 0, 0,
      nullptr, 0.f, qhi, Wl, logits);

  // 5) softmax over regions + v_emb = sum_k att_k * gs[n,k,:]
  softmax_vemb<<<N_TOT, 256, 0, stream>>>(logits, gs, vemb);

  // 6) out = relu(v_emb @ Wv + bv) * q_repr
  gemm_wmma_f32<EPI_BIAS_RELU_MUL><<<dim3(D_DIM / 64, N_TOT / 128, 1), blk, 0, stream>>>(
      vemb, Wv, (float*)d_out, N_TOT, D_DIM, D_DIM, 0, 0, 0,
      bv, 1.f, qrep, nullptr, nullptr);
}